// R_roberta_70360154243670
// MI455X (gfx1250) — compile-verified
//
#include <hip/hip_runtime.h>
#include <hip/hip_bf16.h>

typedef __attribute__((ext_vector_type(2))) float v2f;
typedef __attribute__((ext_vector_type(8))) float v8f;

#define BB 64
#define TT 512
#define HH 768
#define LL 30
#define CAT (3 * HH)  // 2304

// ---------------------------------------------------------------------------
// Kernel 1: span-compaction + mean pool + tanh.
// One block per batch row. Thread 0 does the sequential run-labeling scan
// (T=512 scalar steps), emitting compact token-index lists for span 1 and
// span 2. Then 256 threads mean-pool ONLY those rows (instead of the 100 MB
// full einsum) and apply tanh.
// ---------------------------------------------------------------------------
__global__ __launch_bounds__(256) void span_tanh_kernel(
    const float* __restrict__ h, const int* __restrict__ ent,
    float* __restrict__ t1, float* __restrict__ t2) {
  __shared__ int toks1[TT];
  __shared__ int toks2[TT];
  __shared__ int s_cnt1, s_cnt2;
  const int b = blockIdx.x;

  if (threadIdx.x == 0) {
    int c1 = 0, c2 = 0, prev = 0, run = 0;
    for (int t = 0; t < TT; ++t) {
      int e = ent[b * TT + t];
      if (e == 1 && prev == 0) run++;
      if (e == 1) {
        if (run == 1)      toks1[c1++] = t;
        else if (run == 2) toks2[c2++] = t;
      }
      prev = e;
    }
    s_cnt1 = c1;
    s_cnt2 = c2;
  }
  __syncthreads();

  const int c1 = s_cnt1, c2 = s_cnt2;
  const float inv1 = 1.0f / (float)(c1 > 1 ? c1 : 1);
  const float inv2 = 1.0f / (float)(c2 > 1 ? c2 : 1);
  const float* hb = h + (size_t)b * TT * HH;

  for (int d = threadIdx.x; d < HH; d += 256) {
    float s1 = 0.0f, s2 = 0.0f;
    for (int i = 0; i < c1; ++i) s1 += hb[(size_t)toks1[i] * HH + d];
    for (int i = 0; i < c2; ++i) s2 += hb[(size_t)toks2[i] * HH + d];
    t1[b * HH + d] = tanhf(s1 * inv1);
    t2[b * HH + d] = tanhf(s2 * inv2);
  }
}

// ---------------------------------------------------------------------------
// Kernel 2: fused triple GEMM on the f32 WMMA path.
//   Y[:,    0: 768] = pooled    @ W_cls + b_cls
//   Y[:,  768:1536] = tanh(e1)  @ W_e1  + b_e1
//   Y[:, 1536:2304] = tanh(e2)  @ W_e2  + b_e2
// 576 16x16 tiles total; 8 waves/block -> 72 blocks. K loop of 4 using
// V_WMMA_F32_16X16X4_F32; bias folded into the C accumulator init.
// ---------------------------------------------------------------------------
__global__ __launch_bounds__(256) void gemm3_kernel(
    const float* __restrict__ pooled,
    const float* __restrict__ t1, const float* __restrict__ t2,
    const float* __restrict__ Wcls, const float* __restrict__ bcls,
    const float* __restrict__ We1,  const float* __restrict__ be1,
    const float* __restrict__ We2,  const float* __restrict__ be2,
    float* __restrict__ Y) {
  const int wave = threadIdx.x >> 5;
  const int lane = threadIdx.x & 31;
  const int g = blockIdx.x * 8 + wave;   // tile id, 0..575
  const int mTile = g / 144;             // 4 M-tiles (64 rows)
  const int nt = g % 144;
  const int seg = nt / 48;               // which of the three GEMMs
  const int nTile = nt % 48;             // 48 N-tiles (768 cols)
  const int m0 = mTile * 16, n0 = nTile * 16;

  const float* A    = (seg == 0) ? pooled : (seg == 1 ? t1 : t2);
  const float* W    = (seg == 0) ? Wcls   : (seg == 1 ? We1 : We2);
  const float* bias = (seg == 0) ? bcls   : (seg == 1 ? be1 : be2);

  const int row = lane & 15;             // A: M within tile
  const int kq  = (lane >> 4) << 1;      // A/B: K pair base (0 or 2)
  const int col = lane & 15;             // B/C: N within tile

  const float bv = bias[n0 + col];
  v8f c;
#pragma unroll
  for (int r = 0; r < 8; ++r) c[r] = bv;

  const float* arow = A + (size_t)(m0 + row) * HH;
  for (int k0 = 0; k0 < HH; k0 += 4) {
    v2f a, bm;
    a.x  = arow[k0 + kq];
    a.y  = arow[k0 + kq + 1];
    bm.x = W[(size_t)(k0 + kq)     * HH + n0 + col];
    bm.y = W[(size_t)(k0 + kq + 1) * HH + n0 + col];
    c = __builtin_amdgcn_wmma_f32_16x16x4_f32(false, a, false, bm,
                                              (short)0, c, false, false);
  }

  const int mhalf = (lane >> 4) << 3;    // 0 or 8
#pragma unroll
  for (int r = 0; r < 8; ++r) {
    const int m = m0 + r + mhalf;
    Y[(size_t)m * CAT + seg * HH + n0 + col] = c[r];
  }
}

// ---------------------------------------------------------------------------
// Kernel 3: logits = Y @ W_cat + b_cat (64x2304x30, N padded to 32), then
// softmax -> probs (output), then log_softmax(probs) -> CE loss.
// Single block, 8 waves = 8 tiles (4 M-tiles x 2 N-tiles), K = 2304.
// Padded N columns accumulate exactly zero (zero B, zero bias) via
// branch-free selects so EXEC stays all-ones around the WMMA ops.
// ---------------------------------------------------------------------------
__global__ __launch_bounds__(256) void head_kernel(
    const float* __restrict__ Y, const float* __restrict__ Wcat,
    const float* __restrict__ bcat, const int* __restrict__ labels,
    float* __restrict__ out) {
  __shared__ float lg[BB][32];
  __shared__ float lred[BB];

  const int wave = threadIdx.x >> 5;
  const int lane = threadIdx.x & 31;
  const int mTile = wave >> 1;           // 0..3
  const int nTile = wave & 1;            // 0..1
  const int m0 = mTile * 16, n0 = nTile * 16;
  const int row = lane & 15;
  const int kq  = (lane >> 4) << 1;
  const int col = lane & 15;
  const int n = n0 + col;
  const bool nvalid = (n < LL);

  const float bv = nvalid ? bcat[n] : 0.0f;
  v8f c;
#pragma unroll
  for (int r = 0; r < 8; ++r) c[r] = bv;

  const float* arow = Y + (size_t)(m0 + row) * CAT;
  const int nsafe = nvalid ? n : 0;      // clamp address, zero the value
  for (int k0 = 0; k0 < CAT; k0 += 4) {
    v2f a, bm;
    a.x = arow[k0 + kq];
    a.y = arow[k0 + kq + 1];
    float w0 = Wcat[(size_t)(k0 + kq)     * LL + nsafe];
    float w1 = Wcat[(size_t)(k0 + kq + 1) * LL + nsafe];
    bm.x = nvalid ? w0 : 0.0f;
    bm.y = nvalid ? w1 : 0.0f;
    c = __builtin_amdgcn_wmma_f32_16x16x4_f32(false, a, false, bm,
                                              (short)0, c, false, false);
  }

  const int mhalf = (lane >> 4) << 3;
#pragma unroll
  for (int r = 0; r < 8; ++r) lg[m0 + r + mhalf][n] = c[r];
  __syncthreads();

  const int tid = threadIdx.x;
  if (tid < BB) {
    float mx = -3.0e38f;
    for (int j = 0; j < LL; ++j) mx = fmaxf(mx, lg[tid][j]);
    float e[LL];
    float s = 0.0f;
    for (int j = 0; j < LL; ++j) { e[j] = __expf(lg[tid][j] - mx); s += e[j]; }
    const float inv = 1.0f / s;
    float p[LL];
    float mx2 = -3.0e38f;
    for (int j = 0; j < LL; ++j) {
      p[j] = e[j] * inv;
      out[tid * LL + j] = p[j];          // probs output
      mx2 = fmaxf(mx2, p[j]);
    }
    float s2 = 0.0f;
    for (int j = 0; j < LL; ++j) s2 += __expf(p[j] - mx2);
    const int lab = labels[tid];
    const float logp = p[lab] - mx2 - __logf(s2);
    lred[tid] = -logp;
  }
  __syncthreads();
  if (tid == 0) {
    float acc = 0.0f;
    for (int i = 0; i < BB; ++i) acc += lred[i];
    out[BB * LL] = acc / (float)BB;      // scalar loss appended
  }
}

// ---------------------------------------------------------------------------
extern "C" void kernel_launch(void* const* d_in, const int* in_sizes, int n_in,
                              void* d_out, int out_size, void* d_ws,
                              size_t ws_size, hipStream_t stream) {
  const float* h      = (const float*)d_in[0];   // [64,512,768]
  const float* pooled = (const float*)d_in[1];   // [64,768]
  const int*   ent    = (const int*)  d_in[2];   // [64,512]
  const int*   labels = (const int*)  d_in[3];   // [64]
  const float* Wcls   = (const float*)d_in[4];
  const float* bcls   = (const float*)d_in[5];
  const float* We1    = (const float*)d_in[6];
  const float* be1    = (const float*)d_in[7];
  const float* We2    = (const float*)d_in[8];
  const float* be2    = (const float*)d_in[9];
  const float* Wcat   = (const float*)d_in[10];  // [2304,30]
  const float* bcat   = (const float*)d_in[11];  // [30]
  float* out = (float*)d_out;                    // [64*30 probs] + [1 loss]

  // Workspace layout (all fp32):
  float* t1 = (float*)d_ws;                      // [64,768]
  float* t2 = t1 + (size_t)BB * HH;              // [64,768]
  float* Yc = t2 + (size_t)BB * HH;              // [64,2304]

  span_tanh_kernel<<<BB, 256, 0, stream>>>(h, ent, t1, t2);
  gemm3_kernel<<<72, 256, 0, stream>>>(pooled, t1, t2, Wcls, bcls, We1, be1,
                                       We2, be2, Yc);
  head_kernel<<<1, 256, 0, stream>>>(Yc, Wcat, bcat, labels, out);
}